// JacobianRegularizer_41944650613016
// MI455X (gfx1250) — compile-verified
//
#include <hip/hip_runtime.h>
#include <math.h>

typedef __attribute__((ext_vector_type(2))) float v2f;
typedef __attribute__((ext_vector_type(4))) float v4f;
typedef __attribute__((ext_vector_type(8))) float v8f;

#define N_DIM 8192
#define M_DIM 8192
#define GRID_MAIN 1024
#define TPB 256   // 8 waves of 32 (wave32)

// ws layout (floats): [0, 8192) = d2 ; [8192, 8192+GRID_MAIN) = block partials

// ---------------- Kernel 1: softmax -> d^2 (tiny, deterministic) ----------------
__global__ void jacreg_softmax_d2(const float* __restrict__ x, float* __restrict__ d2) {
    __shared__ float red[TPB];
    const int tid = threadIdx.x;

    // pass 1: max
    float m = -3.402823466e38f;
    for (int i = tid; i < N_DIM; i += TPB) m = fmaxf(m, x[i]);
    red[tid] = m;
    __syncthreads();
    for (int s = TPB / 2; s > 0; s >>= 1) {
        if (tid < s) red[tid] = fmaxf(red[tid], red[tid + s]);
        __syncthreads();
    }
    const float xmax = red[0];
    __syncthreads();

    // pass 2: sum of exp (fixed order per thread, fixed tree -> deterministic)
    float sum = 0.0f;
    for (int i = tid; i < N_DIM; i += TPB) sum += expf(x[i] - xmax);
    red[tid] = sum;
    __syncthreads();
    for (int s = TPB / 2; s > 0; s >>= 1) {
        if (tid < s) red[tid] += red[tid + s];
        __syncthreads();
    }
    const float inv = 1.0f / red[0];
    __syncthreads();

    // pass 3: d2[i] = (sx*(1-sx))^2
    for (int i = tid; i < N_DIM; i += TPB) {
        float sx = expf(x[i] - xmax) * inv;
        float d  = sx * (1.0f - sx);
        d2[i] = d * d;
    }
}

// ---------------- Kernel 2: streaming weighted sum over w (bandwidth-bound) ----------------
// Each block strides over rows. Within a block, wave v owns columns
// [v*1024, v*1024+1024): 8 iterations of 32 lanes x float4 = 128 floats, coalesced
// 512B-per-wave B128 NT loads. Row weight d2[r] is block-uniform (SGPR) and is
// applied once per row. Final cross-lane fold uses V_WMMA_F32_16X16X4_F32.
__global__ void jacreg_main(const float* __restrict__ w,
                            const float* __restrict__ d2,
                            float* __restrict__ partials) {
    const int tid  = threadIdx.x;
    const int lane = tid & 31;
    const int wave = tid >> 5;       // 0..7

    float acc = 0.0f;                // per-lane weighted accumulator

    for (int r = blockIdx.x; r < N_DIM; r += gridDim.x) {
        const float s = d2[r];       // uniform across block -> scalar load
        const float* base = w + (size_t)r * M_DIM + wave * 1024 + lane * 4;

        float rp0 = 0.0f, rp1 = 0.0f, rp2 = 0.0f, rp3 = 0.0f;
#pragma unroll
        for (int t = 0; t < 8; ++t) {
            v4f v = __builtin_nontemporal_load((const v4f*)(base + t * 128));
            rp0 = fmaf(v.x, v.x, rp0);
            rp1 = fmaf(v.y, v.y, rp1);
            rp2 = fmaf(v.z, v.z, rp2);
            rp3 = fmaf(v.w, v.w, rp3);
        }
        acc = fmaf(s, (rp0 + rp1) + (rp2 + rp3), acc);
    }

    // Cross-lane reduce via WMMA: A = 16x4 f32 (slot0 = acc, slot1 = 0), B = ones.
    // D[i][j] = acc(lane i) + acc(lane i+16) for every column j.
    // Lanes 0..15 hold D rows 0..7 (col = lane), lanes 16..31 hold rows 8..15.
    v2f a; a.x = acc;  a.y = 0.0f;
    v2f b; b.x = 1.0f; b.y = 1.0f;
    v8f c = {};
    c = __builtin_amdgcn_wmma_f32_16x16x4_f32(
        /*neg_a=*/false, a, /*neg_b=*/false, b,
        /*c_mod=*/(short)0, c, /*reuse_a=*/false, /*reuse_b=*/false);

    float p = ((c[0] + c[1]) + (c[2] + c[3])) + ((c[4] + c[5]) + (c[6] + c[7]));
    // lanes 0..15 all hold sum(rows 0..7); lanes 16..31 all hold sum(rows 8..15)
    float q = __shfl_down(p, 16, 32);

    __shared__ float wsum[TPB / 32];
    if (lane == 0) wsum[wave] = p + q;   // wave total
    __syncthreads();

    if (tid == 0) {
        float t = 0.0f;
        for (int i = 0; i < TPB / 32; ++i) t += wsum[i];
        partials[blockIdx.x] = t;
    }
}

// ---------------- Kernel 3: reduce partials + sqrt ----------------
__global__ void jacreg_final(const float* __restrict__ partials, float* __restrict__ out) {
    __shared__ float red[TPB];
    const int tid = threadIdx.x;
    float s = 0.0f;
    for (int i = tid; i < GRID_MAIN; i += TPB) s += partials[i];
    red[tid] = s;
    __syncthreads();
    for (int st = TPB / 2; st > 0; st >>= 1) {
        if (tid < st) red[tid] += red[tid + st];
        __syncthreads();
    }
    if (tid == 0) out[0] = sqrtf(red[0]);
}

extern "C" void kernel_launch(void* const* d_in, const int* in_sizes, int n_in,
                              void* d_out, int out_size, void* d_ws, size_t ws_size,
                              hipStream_t stream) {
    const float* x = (const float*)d_in[0];   // [8192]
    const float* w = (const float*)d_in[1];   // [8192, 8192]
    float* out = (float*)d_out;               // scalar
    float* ws  = (float*)d_ws;
    float* d2       = ws;                     // 8192 floats
    float* partials = ws + N_DIM;             // GRID_MAIN floats

    jacreg_softmax_d2<<<1, TPB, 0, stream>>>(x, d2);
    jacreg_main<<<GRID_MAIN, TPB, 0, stream>>>(w, d2, partials);
    jacreg_final<<<1, TPB, 0, stream>>>(partials, out);
}